// GAT_71399536328825
// MI455X (gfx1250) — compile-verified
//
#include <hip/hip_runtime.h>
#include <hip/hip_bf16.h>

typedef __attribute__((ext_vector_type(2))) float v2f;
typedef __attribute__((ext_vector_type(8))) float v8f;

#define NEG_SLOPE 0.2f
#define KPAD_MAX 12   // max padded K (layer 0: Fin=10 -> 12)
#define NPAD 16       // padded N dimension for W staging

// ---------- helpers: order-preserving float<->uint encoding for atomicMax ----------
__device__ __forceinline__ unsigned enc_f32(float f) {
  unsigned u = __float_as_uint(f);
  return (u >> 31) ? ~u : (u | 0x80000000u);
}
__device__ __forceinline__ float dec_f32(unsigned u) {
  return (u >> 31) ? __uint_as_float(u & 0x7fffffffu) : __uint_as_float(~u);
}

// ---------- K0: build int32 edge lists with self-loops appended ----------
__global__ void build_edges_kernel(const int* __restrict__ eidx, int* __restrict__ src,
                                   int* __restrict__ dst, long long E, int N) {
  long long i = blockIdx.x * (long long)blockDim.x + threadIdx.x;
  long long Etot = E + (long long)N;
  if (i >= Etot) return;
  if (i < E) {
    src[i] = eidx[i];
    dst[i] = eidx[E + i];
  } else {
    int v = (int)(i - E);
    src[i] = v;
    dst[i] = v;
  }
}

// ---------- pad layer-0 input: x[N,10] -> xpad[N,12] (zero padded) ----------
__global__ void pad_x_kernel(const float* __restrict__ x, float* __restrict__ xpad, int N) {
  long long i = blockIdx.x * (long long)blockDim.x + threadIdx.x;
  long long tot = (long long)N * KPAD_MAX;
  if (i >= tot) return;
  int row = (int)(i / KPAD_MAX);
  int k = (int)(i % KPAD_MAX);
  xpad[i] = (k < 10) ? x[(long long)row * 10 + k] : 0.f;
}

// ---------- expand W[Fin,Fout] -> Wpad[12,16] (zero padded) ----------
__global__ void pad_w_kernel(const float* __restrict__ W, float* __restrict__ Wpad,
                             int Fin, int Fout) {
  int i = threadIdx.x;                 // one block of 256 covers 12*16=192
  if (i >= KPAD_MAX * NPAD) return;
  int k = i / NPAD, n = i % NPAD;
  Wpad[i] = (k < Fin && n < Fout) ? W[k * Fout + n] : 0.f;
}

// ---------- K1: h = (relu?)x @ W via V_WMMA_F32_16X16X4_F32, one wave = 16 rows ----------
// x has row stride KT*4 (zero padded), Wpad is [12,16] zero padded -> all loads unconditional.
template <int KT, int FOUT, bool RELU>
__global__ void gemm_wmma_kernel(const float* __restrict__ x, const float* __restrict__ Wpad,
                                 float* __restrict__ h, int N) {
  const int XS = KT * 4;               // padded input row stride
  int lane = threadIdx.x & 31;
  int wave = threadIdx.x >> 5;
  int rowBase = (blockIdx.x * (blockDim.x >> 5) + wave) * 16;
  if (rowBase >= N) return;            // wave-uniform: EXEC stays all-ones for WMMA
  int hi = lane >> 4;                  // 0 or 1
  int lo = lane & 15;
  int row = rowBase + lo;

  v8f c = {};
#pragma unroll
  for (int kt = 0; kt < KT; ++kt) {
    int k0 = kt * 4 + hi * 2;
    // A[M=lo][K=k0..k0+1]: one aligned 8-byte load, no predication
    v2f a = *(const v2f*)(x + (long long)row * XS + k0);
    if (RELU) { a[0] = fmaxf(a[0], 0.f); a[1] = fmaxf(a[1], 0.f); }
    v2f b;
    b[0] = Wpad[k0 * NPAD + lo];
    b[1] = Wpad[(k0 + 1) * NPAD + lo];
    c = __builtin_amdgcn_wmma_f32_16x16x4_f32(false, a, false, b, (short)0, c, false, false);
  }
  // C layout: lane L, reg r -> element [rowBase + r + 8*hi][lo]
  if (lo < FOUT) {
#pragma unroll
    for (int r = 0; r < 8; ++r) {
      h[(long long)(rowBase + r + 8 * hi) * FOUT + lo] = c[r];
    }
  }
}

// ---------- K2: per-node alphas + init m/denom + acc = bias (stride S, zero pad) ----------
template <int F, int S>
__global__ void node_prep_kernel(const float* __restrict__ h, const float* __restrict__ asrc,
                                 const float* __restrict__ adst, const float* __restrict__ bias,
                                 float* __restrict__ alphas, float* __restrict__ alphad,
                                 unsigned* __restrict__ menc, float* __restrict__ denom,
                                 float* __restrict__ acc, int N) {
  int i = blockIdx.x * blockDim.x + threadIdx.x;
  if (i >= N) return;
  float s = 0.f, t = 0.f;
#pragma unroll
  for (int f = 0; f < F; ++f) {
    float hv = h[(long long)i * F + f];
    s += hv * asrc[f];
    t += hv * adst[f];
    acc[(long long)i * S + f] = bias[f];
  }
#pragma unroll
  for (int f = F; f < S; ++f) acc[(long long)i * S + f] = 0.f;  // keep GEMM padding zero
  alphas[i] = s;
  alphad[i] = t;
  menc[i] = 0u;      // encodes below any real value (every dst has a self-loop)
  denom[i] = 0.f;
}

// ---------- K3: e = leakyrelu(as[src]+ad[dst]); segment-max into menc[dst] ----------
__global__ void edge_max_kernel(const int* __restrict__ src, const int* __restrict__ dst,
                                const float* __restrict__ as, const float* __restrict__ ad,
                                float* __restrict__ e, unsigned* __restrict__ menc,
                                long long Etot) {
  long long i = blockIdx.x * (long long)blockDim.x + threadIdx.x;
  if (i >= Etot) return;
  int s = src[i], d = dst[i];
  float v = as[s] + ad[d];
  v = (v >= 0.f) ? v : NEG_SLOPE * v;
  e[i] = v;
  atomicMax(&menc[d], enc_f32(v));
}

// ---------- K4: ex = exp(e - m[dst]); segment-sum into denom[dst] ----------
__global__ void edge_exp_kernel(const int* __restrict__ dst, float* __restrict__ e,
                                const unsigned* __restrict__ menc, float* __restrict__ denom,
                                long long Etot) {
  long long i = blockIdx.x * (long long)blockDim.x + threadIdx.x;
  if (i >= Etot) return;
  int d = dst[i];
  float ex = expf(e[i] - dec_f32(menc[d]));
  e[i] = ex;
  atomicAdd(&denom[d], ex);
}

// ---------- K5: acc[dst] += (ex/denom[dst]) * h[src]  (acc stride S, h stride F) ----------
template <int F, int S>
__global__ void edge_scatter_kernel(const int* __restrict__ src, const int* __restrict__ dst,
                                    const float* __restrict__ e, const float* __restrict__ denom,
                                    const float* __restrict__ h, float* __restrict__ acc,
                                    long long Etot) {
  long long i = blockIdx.x * (long long)blockDim.x + threadIdx.x;
  if (i >= Etot) return;
  int s = src[i], d = dst[i];
  float w = e[i] / denom[d];
#pragma unroll
  for (int f = 0; f < F; ++f) {
    atomicAdd(&acc[(long long)d * S + f], w * h[(long long)s * F + f]);
  }
}

// =====================================================================================
extern "C" void kernel_launch(void* const* d_in, const int* in_sizes, int n_in,
                              void* d_out, int out_size, void* d_ws, size_t ws_size,
                              hipStream_t stream) {
  (void)n_in; (void)ws_size; (void)out_size;
  const int NLAYERS = 10;

  const float* x0  = (const float*)d_in[0];
  const int* eidx  = (const int*)d_in[1];  // [2, E] int32 (harness integer convention)
  const float* Ws[NLAYERS];
  const float* Asrc[NLAYERS];
  const float* Adst[NLAYERS];
  const float* Bias[NLAYERS];
  for (int l = 0; l < NLAYERS; ++l) {
    Ws[l]   = (const float*)d_in[2 + l];
    Asrc[l] = (const float*)d_in[12 + l];
    Adst[l] = (const float*)d_in[22 + l];
    Bias[l] = (const float*)d_in[32 + l];
  }

  const int Fin0 = 10;
  const int N = in_sizes[0] / Fin0;            // 100000
  const long long E = in_sizes[1] / 2;         // 3200000
  const long long Etot = E + (long long)N;     // + self-loops

  // ---- carve workspace (256B aligned slices) ----
  char* p = (char*)d_ws;
  auto carve = [&](size_t bytes) -> void* {
    void* r = (void*)p;
    p += (bytes + 255) & ~(size_t)255;
    return r;
  };
  int*      src32  = (int*)carve(Etot * sizeof(int));
  int*      dst32  = (int*)carve(Etot * sizeof(int));
  float*    ebuf   = (float*)carve(Etot * sizeof(float));          // e, then ex
  float*    xpad   = (float*)carve((size_t)N * KPAD_MAX * sizeof(float));  // layer-0 input, stride 12
  float*    accA   = (float*)carve((size_t)N * 8 * sizeof(float)); // inter-layer, stride 8
  float*    accB   = (float*)carve((size_t)N * 8 * sizeof(float));
  float*    hbuf   = (float*)carve((size_t)N * 5 * sizeof(float)); // compact h
  float*    Wpad   = (float*)carve((size_t)KPAD_MAX * NPAD * sizeof(float));
  float*    alphas = (float*)carve((size_t)N * sizeof(float));
  float*    alphad = (float*)carve((size_t)N * sizeof(float));
  unsigned* menc   = (unsigned*)carve((size_t)N * sizeof(unsigned));
  float*    denom  = (float*)carve((size_t)N * sizeof(float));

  const int BLK = 256;
  unsigned gridE = (unsigned)((Etot + BLK - 1) / BLK);
  unsigned gridN = (unsigned)((N + BLK - 1) / BLK);
  unsigned gridP = (unsigned)(((long long)N * KPAD_MAX + BLK - 1) / BLK);
  unsigned gridG = (unsigned)((N + 16 * 8 - 1) / (16 * 8));  // 8 waves/block, 16 rows/wave

  build_edges_kernel<<<gridE, BLK, 0, stream>>>(eidx, src32, dst32, E, N);
  pad_x_kernel<<<gridP, BLK, 0, stream>>>(x0, xpad, N);

  const float* cur = xpad;
  for (int l = 0; l < NLAYERS; ++l) {
    int Fin  = (l == 0) ? 10 : 5;
    int Fout = (l == NLAYERS - 1) ? 1 : 5;
    float* acc = (l == NLAYERS - 1) ? (float*)d_out : ((l & 1) ? accB : accA);

    pad_w_kernel<<<1, BLK, 0, stream>>>(Ws[l], Wpad, Fin, Fout);

    if (l == 0) {
      gemm_wmma_kernel<3, 5, false><<<gridG, BLK, 0, stream>>>(cur, Wpad, hbuf, N);
    } else if (l < NLAYERS - 1) {
      gemm_wmma_kernel<2, 5, true><<<gridG, BLK, 0, stream>>>(cur, Wpad, hbuf, N);
    } else {
      gemm_wmma_kernel<2, 1, true><<<gridG, BLK, 0, stream>>>(cur, Wpad, hbuf, N);
    }

    if (l < NLAYERS - 1) {
      node_prep_kernel<5, 8><<<gridN, BLK, 0, stream>>>(hbuf, Asrc[l], Adst[l], Bias[l],
                                                        alphas, alphad, menc, denom, acc, N);
    } else {
      node_prep_kernel<1, 1><<<gridN, BLK, 0, stream>>>(hbuf, Asrc[l], Adst[l], Bias[l],
                                                        alphas, alphad, menc, denom, acc, N);
    }

    edge_max_kernel<<<gridE, BLK, 0, stream>>>(src32, dst32, alphas, alphad, ebuf, menc, Etot);
    edge_exp_kernel<<<gridE, BLK, 0, stream>>>(dst32, ebuf, menc, denom, Etot);

    if (l < NLAYERS - 1) {
      edge_scatter_kernel<5, 8><<<gridE, BLK, 0, stream>>>(src32, dst32, ebuf, denom, hbuf,
                                                           acc, Etot);
    } else {
      edge_scatter_kernel<1, 1><<<gridE, BLK, 0, stream>>>(src32, dst32, ebuf, denom, hbuf,
                                                           acc, Etot);
    }
    cur = acc;
  }
}